// DRewGIN_53609781789207
// MI455X (gfx1250) — compile-verified
//
#include <hip/hip_runtime.h>
#include <hip/hip_bf16.h>

typedef __attribute__((ext_vector_type(2))) float v2f;
typedef __attribute__((ext_vector_type(8))) float v8f;

#define DIM 128

// ---------------------------------------------------------------------------
// Epilogue helper: apply bias+relu (+scale or +accumulate) and store 8 rows
// of one 16x16 tile. GUARD=false is the straight-line full-tile path (no
// per-element exec-mask churn); GUARD=true handles a ragged last tile.
// ---------------------------------------------------------------------------
template <int ACC, bool GUARD>
__device__ __forceinline__ void DRewGIN_epilogue(
    const v8f& c, float* __restrict__ Out, const float* __restrict__ bias,
    int rbase, int col, float scale, int Nrows)
{
    const float bv = bias[col];
    #pragma unroll
    for (int i = 0; i < 8; ++i) {
        const int r = rbase + i;
        if (!GUARD || r < Nrows) {
            float v = fmaxf(c[i] + bv, 0.0f);
            float* o = Out + (size_t)r * DIM + col;
            if (ACC) *o += v;
            else     *o  = scale * v;
        }
    }
}

// ---------------------------------------------------------------------------
// GEMM: Out[N x 128] = op( relu( X[N x 128] @ W[128 x 128] + b ) )
//   ACC == 0 : Out  = (1 + eps[epsIdx]) * relu(...)
//   ACC == 1 : Out += relu(...)
// One wave computes a 16x64 output slab as 4 independent 16x16 WMMA
// accumulators (breaks the WMMA->WMMA RAW chain; A fragment amortized over 4
// WMMAs). 2 waves/block cover one 16-row x 128-col strip.
// ---------------------------------------------------------------------------
template <int ACC>
__global__ __launch_bounds__(64) void DRewGIN_gemm_relu(
    const float* __restrict__ X, const float* __restrict__ W,
    const float* __restrict__ bias, float* __restrict__ Out,
    const float* __restrict__ epsArr, int epsIdx, int Nrows)
{
    const int lane  = threadIdx.x & 31;
    const int wave  = threadIdx.x >> 5;      // 0..1 -> column half (64 cols)
    const int m     = lane & 15;             // row index within half-wave
    const int khalf = lane >> 4;             // 0: K pair {0,1}; 1: K pair {2,3}
    const int rowT  = blockIdx.x;            // 16-row strip

    int row = rowT * 16 + m;
    if (row >= Nrows) row = Nrows - 1;       // clamp loads (stores predicated)
    const int col0 = wave * 64 + m;          // tile j covers col0 + 16*j

    const float* __restrict__ xrow = X + (size_t)row * DIM;
    const float* __restrict__ wcol = W + col0;

    v8f c0 = {}, c1 = {}, c2 = {}, c3 = {};
    #pragma unroll 2
    for (int k0 = 0; k0 < DIM; k0 += 4) {
        const int ka = k0 + khalf * 2;       // ISA A layout: v0=K{0|2}, v1=K{1|3}
        v2f a;
        a.x = xrow[ka];
        a.y = xrow[ka + 1];
        const float* w0 = wcol + (size_t)ka * DIM;
        const float* w1 = w0 + DIM;
        v2f b0, b1, b2, b3;
        b0.x = w0[0];  b0.y = w1[0];
        b1.x = w0[16]; b1.y = w1[16];
        b2.x = w0[32]; b2.y = w1[32];
        b3.x = w0[48]; b3.y = w1[48];
        c0 = __builtin_amdgcn_wmma_f32_16x16x4_f32(false, a, false, b0, (short)0, c0, false, false);
        c1 = __builtin_amdgcn_wmma_f32_16x16x4_f32(false, a, false, b1, (short)0, c1, false, false);
        c2 = __builtin_amdgcn_wmma_f32_16x16x4_f32(false, a, false, b2, (short)0, c2, false, false);
        c3 = __builtin_amdgcn_wmma_f32_16x16x4_f32(false, a, false, b3, (short)0, c3, false, false);
    }

    const float scale = 1.0f + epsArr[epsIdx];
    const int rbase = rowT * 16 + khalf * 8; // C/D layout: VGPR i -> M = i + 8*khalf

    if (rowT * 16 + 16 <= Nrows) {
        // Uniform full-tile fast path: straight-line stores, no exec masking.
        DRewGIN_epilogue<ACC, false>(c0, Out, bias, rbase, col0,      scale, Nrows);
        DRewGIN_epilogue<ACC, false>(c1, Out, bias, rbase, col0 + 16, scale, Nrows);
        DRewGIN_epilogue<ACC, false>(c2, Out, bias, rbase, col0 + 32, scale, Nrows);
        DRewGIN_epilogue<ACC, false>(c3, Out, bias, rbase, col0 + 48, scale, Nrows);
    } else {
        DRewGIN_epilogue<ACC, true>(c0, Out, bias, rbase, col0,      scale, Nrows);
        DRewGIN_epilogue<ACC, true>(c1, Out, bias, rbase, col0 + 16, scale, Nrows);
        DRewGIN_epilogue<ACC, true>(c2, Out, bias, rbase, col0 + 32, scale, Nrows);
        DRewGIN_epilogue<ACC, true>(c3, Out, bias, rbase, col0 + 48, scale, Nrows);
    }
}

// ---------------------------------------------------------------------------
// Zero a float buffer (float4-vectorized).
// ---------------------------------------------------------------------------
__global__ __launch_bounds__(256) void DRewGIN_zero(float4* __restrict__ p, int n4)
{
    int i = blockIdx.x * blockDim.x + threadIdx.x;
    if (i < n4) p[i] = make_float4(0.f, 0.f, 0.f, 0.f);
}

// ---------------------------------------------------------------------------
// Masked gather + scatter-add: for edges with attr==k,
//   agg[dst[e], :] += xsrc[src[e], :]
// One wave per edge: 32 lanes x float4 = the full 128-float row (512B
// coalesced gather from L2-resident x), fp32 atomics into L2 atomic units.
// ---------------------------------------------------------------------------
__global__ __launch_bounds__(256) void DRewGIN_scatter(
    const float* __restrict__ xsrc, const int* __restrict__ src,
    const int* __restrict__ dst, const int* __restrict__ attr,
    int k, float* __restrict__ agg, int E)
{
    const int e = blockIdx.x * 8 + (threadIdx.x >> 5);
    if (e >= E) return;
    if (attr[e] != k) return;
    const int lane = threadIdx.x & 31;
    const int s = src[e];
    const int d = dst[e];
    const float4 v = *((const float4*)(xsrc + (size_t)s * DIM) + lane);
    float* o = agg + (size_t)d * DIM + lane * 4;
    atomicAdd(o + 0, v.x);
    atomicAdd(o + 1, v.y);
    atomicAdd(o + 2, v.z);
    atomicAdd(o + 3, v.w);
}

// ---------------------------------------------------------------------------
// Residual: xnext = xt + relu(out)   (float4-vectorized)
// ---------------------------------------------------------------------------
__global__ __launch_bounds__(256) void DRewGIN_residual(
    const float4* __restrict__ xt, const float4* __restrict__ out,
    float4* __restrict__ xnext, int n4)
{
    int i = blockIdx.x * blockDim.x + threadIdx.x;
    if (i < n4) {
        float4 a = xt[i];
        float4 o = out[i];
        float4 r;
        r.x = a.x + fmaxf(o.x, 0.0f);
        r.y = a.y + fmaxf(o.y, 0.0f);
        r.z = a.z + fmaxf(o.z, 0.0f);
        r.w = a.w + fmaxf(o.w, 0.0f);
        xnext[i] = r;
    }
}

extern "C" void kernel_launch(void* const* d_in, const int* in_sizes, int n_in,
                              void* d_out, int out_size, void* d_ws, size_t ws_size,
                              hipStream_t stream)
{
    const float* x    = (const float*)d_in[0];   // [N, 128]
    const int*   ei   = (const int*)  d_in[1];   // [2, E]
    const int*   attr = (const int*)  d_in[2];   // [E]
    const float* Ws   = (const float*)d_in[3];   // [3, 128, 128]
    const float* bs   = (const float*)d_in[4];   // [3, 128]
    const float* Wk   = (const float*)d_in[5];   // [3, 3, 128, 128]
    const float* bk   = (const float*)d_in[6];   // [3, 3, 128]
    const float* eps  = (const float*)d_in[7];   // [3]

    const int N = in_sizes[0] / DIM;
    const int E = in_sizes[2];
    const int* src = ei;
    const int* dst = ei + E;

    const size_t nd = (size_t)N * DIM;
    float* xs1 = (float*)d_ws;
    float* xs2 = xs1 + nd;
    float* agg = xs2 + nd;
    float* out = agg + nd;
    float* xsArr[4] = { (float*)x, xs1, xs2, (float*)d_out };

    const int n4 = (int)(nd / 4);
    const dim3 blkVec(256), blkGemm(64);
    const dim3 gridVec((n4 + 255) / 256);
    const dim3 gridGemm((N + 15) / 16);
    const dim3 gridScat((E + 7) / 8);

    const int L = 3;
    for (int t = 0; t < L; ++t) {
        const float* xt = xsArr[t];
        // out = (1 + eps[t]) * relu(xt @ Ws[t] + bs[t])
        DRewGIN_gemm_relu<0><<<gridGemm, blkGemm, 0, stream>>>(
            xt, Ws + (size_t)t * DIM * DIM, bs + (size_t)t * DIM,
            out, eps, t, N);

        for (int k = 1; k <= t + 1; ++k) {
            const int delay = k - 1;                 // max(k - NU, 0), NU = 1
            const float* xsrc = xsArr[t - delay];
            DRewGIN_zero<<<gridVec, blkVec, 0, stream>>>((float4*)agg, n4);
            DRewGIN_scatter<<<gridScat, blkVec, 0, stream>>>(
                xsrc, src, dst, attr, k, agg, E);
            // out += relu(agg @ Wk[t, k-1] + bk[t, k-1])
            const size_t wofs = ((size_t)t * L + (k - 1)) * DIM * DIM;
            const size_t bofs = ((size_t)t * L + (k - 1)) * DIM;
            DRewGIN_gemm_relu<1><<<gridGemm, blkGemm, 0, stream>>>(
                agg, Wk + wofs, bk + bofs, out, eps, t, N);
        }

        // x_{t+1} = xt + relu(out)
        DRewGIN_residual<<<gridVec, blkVec, 0, stream>>>(
            (const float4*)xt, (const float4*)out, (float4*)xsArr[t + 1], n4);
    }
}